// DGCNN_74414603371141
// MI455X (gfx1250) — compile-verified
//
#include <hip/hip_runtime.h>
#include <math.h>

// Problem constants
#define BB 512
#define CC 64
#define TT 512
#define HH 128
#define KK 16
#define NN (BB*CC)          // 32768
#define EPS_BN 1e-5f
#define APITCH 68           // 64 + 4 pad: rows 16B-aligned, bank-conflict-free

typedef float v2f __attribute__((ext_vector_type(2)));
typedef float v8f __attribute__((ext_vector_type(8)));
typedef int   v4i __attribute__((vector_size(16)));   // builtin's expected pointee

#if __has_builtin(__builtin_amdgcn_global_load_async_to_lds_b128) && \
    __has_builtin(__builtin_amdgcn_s_wait_asynccnt)
#define USE_ASYNC 1
#define NBUF 2
#else
#define USE_ASYNC 0
#define NBUF 1
#endif

// async 16B global -> LDS copy (ASYNCcnt-tracked); no-op stub when unavailable
__device__ __forceinline__ void cp_async16(const float* g, float* l) {
#if USE_ASYNC
  __builtin_amdgcn_global_load_async_to_lds_b128(
      (__attribute__((address_space(1))) v4i*)(uintptr_t)g,
      (__attribute__((address_space(3))) v4i*)l, 0, 0);
#else
  (void)g; (void)l;
#endif
}
__device__ __forceinline__ void wait_async0() {
#if USE_ASYNC
  __builtin_amdgcn_s_wait_asynccnt(0);
#endif
}

// ---------------------------------------------------------------------------
// K1: per-(b,c) inverse row norm:  rinv = 1 / max(||x[b,c,:]||, 1e-12)
__global__ void __launch_bounds__(256) k_rownorm(const float* __restrict__ x,
                                                 float* __restrict__ rinv) {
  int row  = blockIdx.x * 8 + (threadIdx.x >> 5);
  int lane = threadIdx.x & 31;
  const float* p = x + (size_t)row * TT;
  float s = 0.f;
  #pragma unroll
  for (int j = 0; j < TT / 32; ++j) { float v = p[lane + j * 32]; s += v * v; }
  #pragma unroll
  for (int o = 16; o; o >>= 1) s += __shfl_xor(s, o, 32);
  if (lane == 0) rinv[row] = 1.0f / fmaxf(sqrtf(s), 1e-12f);
}

// ---------------------------------------------------------------------------
// K2: materialize nodes = reshape(transpose(x,(0,2,1)), (N,T)) via LDS tiles.
__global__ void __launch_bounds__(256) k_nodes(const float* __restrict__ x,
                                               float* __restrict__ nodes) {
  __shared__ float tile[64][65];
  int b  = blockIdx.x >> 3;
  int t0 = (blockIdx.x & 7) << 6;
  const float* xb = x + (size_t)b * CC * TT;
  for (int i = threadIdx.x; i < 4096; i += 256) {
    int ch = i >> 6, tt = i & 63;
    tile[ch][tt] = xb[ch * TT + t0 + tt];
  }
  __syncthreads();
  float* nb = nodes + (size_t)b * (CC * TT) + (size_t)t0 * 64;
  for (int i = threadIdx.x; i < 4096; i += 256)
    nb[i] = tile[i & 63][i >> 6];
}

// ---------------------------------------------------------------------------
// K3: per-b scaled Gram partial: simpart[b,c,d] = rinv_bc*rinv_bd * <x_bc, x_bd>
__global__ void __launch_bounds__(256) k_simpart(const float* __restrict__ x,
                                                 const float* __restrict__ rinv,
                                                 float* __restrict__ simpart) {
  __shared__ float xs[64][65];
  int b = blockIdx.x;
  const float* xb = x + (size_t)b * CC * TT;
  int c  = threadIdx.x >> 2;
  int dg = threadIdx.x & 3;
  float acc[16];
  #pragma unroll
  for (int i = 0; i < 16; ++i) acc[i] = 0.f;
  for (int kc = 0; kc < 8; ++kc) {
    __syncthreads();
    for (int i = threadIdx.x; i < 4096; i += 256) {
      int ch = i >> 6, j = i & 63;
      xs[ch][j] = xb[ch * TT + (kc << 6) + j];
    }
    __syncthreads();
    for (int j = 0; j < 64; ++j) {
      float av = xs[c][j];
      #pragma unroll
      for (int i = 0; i < 16; ++i) acc[i] += av * xs[dg * 16 + i][j];
    }
  }
  float rc = rinv[b * CC + c];
  float* out = simpart + (size_t)b * 4096 + c * 64 + dg * 16;
  #pragma unroll
  for (int i = 0; i < 16; ++i)
    out[i] = acc[i] * rc * rinv[b * CC + dg * 16 + i];
}

// K4: deterministic reduction over b, plus /B
__global__ void __launch_bounds__(256) k_simreduce(const float* __restrict__ simpart,
                                                   float* __restrict__ sim) {
  int idx = blockIdx.x * 256 + threadIdx.x;   // 0..4095
  float s = 0.f;
  for (int b = 0; b < BB; ++b) s += simpart[(size_t)b * 4096 + idx];
  sim[idx] = s * (1.0f / (float)BB);
}

// ---------------------------------------------------------------------------
// K5: top-16 per row (strict '>' + ascending scan == jax.lax.top_k tie-break),
// degrees, and dense normalized adjacency M[d][c]. One block, 64 threads.
__global__ void __launch_bounds__(64) k_graph(const float* __restrict__ sim,
                                              float* __restrict__ Mmat) {
  __shared__ float simL[64 * 64];
  __shared__ float Ml[64 * 64];
  __shared__ int   indeg[64];
  __shared__ float dinv[64];
  int c = threadIdx.x;
  for (int i = c; i < 4096; i += 64) simL[i] = sim[i];
  indeg[c] = 0;
  __syncthreads();
  unsigned long long taken = 0ull;
  int nbrL[KK];
  const float* row = simL + c * 64;
  for (int k = 0; k < KK; ++k) {
    float best = -3.4e38f; int bi = 0;
    for (int d = 0; d < 64; ++d) {
      bool free = !((taken >> d) & 1ull);
      float v = row[d];
      if (free && v > best) { best = v; bi = d; }
    }
    taken |= (1ull << bi);
    nbrL[k] = bi;
    atomicAdd(&indeg[bi], 1);
  }
  __syncthreads();
  dinv[c] = rsqrtf(1.0f + (float)indeg[c]);   // deg = self-loop + in-degree
  for (int d = 0; d < 64; ++d) Ml[d * 64 + c] = 0.f;
  __syncthreads();
  float dc = dinv[c];
  for (int k = 0; k < KK; ++k) {
    int d = nbrL[k];
    Ml[d * 64 + c] += dinv[d] * dc;           // column c is thread-private: race-free
  }
  Ml[c * 64 + c] += dc * dc;                  // GCN self-loop
  __syncthreads();
  for (int i = c; i < 4096; i += 64) Mmat[i] = Ml[i];
}

// ---------------------------------------------------------------------------
// K6: fused GCN layer core:  Z_b = M @ (A_b @ W) + bias   (per block b),
// plus per-block BN partial sums. Double-buffered ASYNC global->LDS staging
// (GLOBAL_LOAD_ASYNC_TO_LDS_B128, ASYNCcnt) overlapped with WMMA-f32 compute.
__global__ void __launch_bounds__(256)
k_gcn_gemm(const float* __restrict__ Ag, int K,
           const float* __restrict__ Wg,
           const float* __restrict__ Mg,
           const float* __restrict__ bias,
           float* __restrict__ Zg,
           float* __restrict__ psum, float* __restrict__ psqr) {
  __shared__ float AsBuf[NBUF][64 * APITCH]; // A chunk 64x64 (+pad)
  __shared__ float WsBuf[NBUF][64 * 128];    // W chunk 64x128; buf0 reused as P
  __shared__ float sSum[HH], sSqr[HH];

  const int tid   = threadIdx.x;
  const int lane  = tid & 31;
  const int wave  = tid >> 5;
  const int m0    = (wave >> 1) << 4;   // 0,16,32,48
  const int nbase = (wave & 1) << 6;    // 0 or 64
  const int lhalf = lane & 15;
  const int kk    = (lane >> 4) << 1;   // lanes<16 -> K0/K1, lanes>=16 -> K2/K3
  const int b     = blockIdx.x;
  const float* Ab = Ag + (size_t)b * 64 * K;

  if (tid < HH) { sSum[tid] = 0.f; sSqr[tid] = 0.f; }

  v8f acc[4] = {};
  const int nchunk = K >> 6;

#if USE_ASYNC
  // --- async double-buffered pipeline ---------------------------------------
  auto stageA = [&](int kc, int bb) {
    const float* src = Ab + (kc << 6);
    #pragma unroll
    for (int t = 0; t < 4; ++t) {               // 1024 float4s, coalesced
      int j = tid + t * 256;
      int r = j >> 4, c4 = (j & 15) << 2;
      cp_async16(src + (size_t)r * K + c4, &AsBuf[bb][r * APITCH + c4]);
    }
  };
  auto stageW = [&](int kc, int bb) {
    const float* src = Wg + (size_t)(kc << 6) * 128;
    #pragma unroll
    for (int t = 0; t < 8; ++t) {               // 2048 float4s, coalesced
      int j = tid + t * 256;
      int r = j >> 5, c4 = (j & 31) << 2;
      cp_async16(src + r * 128 + c4, &WsBuf[bb][r * 128 + c4]);
    }
  };
  stageA(0, 0); stageW(0, 0);
  for (int kc = 0; kc < nchunk; ++kc) {
    const int cur = kc & 1;
    wait_async0();                 // my outstanding loads for buf 'cur' done
    __syncthreads();               // everyone arrived; buf cur^1 free to fill
    if (kc + 1 < nchunk) { stageA(kc + 1, cur ^ 1); stageW(kc + 1, cur ^ 1); }
    const float* Asb = AsBuf[cur];
    const float* Wsb = WsBuf[cur];
    #pragma unroll 4
    for (int k0 = 0; k0 < 64; k0 += 4) {
      v2f a = *(const v2f*)(Asb + (m0 + lhalf) * APITCH + k0 + kk);  // ds_load_b64
      const float* w0 = Wsb + (k0 + kk) * 128 + nbase + lhalf;
      const float* w1 = Wsb + (k0 + kk + 1) * 128 + nbase + lhalf;
      #pragma unroll
      for (int nt = 0; nt < 4; ++nt) {
        v2f bf; bf.x = w0[nt * 16]; bf.y = w1[nt * 16];
        acc[nt] = __builtin_amdgcn_wmma_f32_16x16x4_f32(
            false, a, false, bf, (short)0, acc[nt], false, false);
      }
    }
  }
#else
  // --- synchronous fallback -------------------------------------------------
  for (int kc = 0; kc < nchunk; ++kc) {
    __syncthreads();
    for (int i = tid; i < 64 * 64; i += 256) {
      int r = i >> 6, cc2 = i & 63;
      AsBuf[0][r * APITCH + cc2] = Ab[(size_t)r * K + (kc << 6) + cc2];
    }
    for (int i = tid; i < 64 * 128; i += 256) {
      int r = i >> 7, cc2 = i & 127;
      WsBuf[0][r * 128 + cc2] = Wg[(size_t)((kc << 6) + r) * 128 + cc2];
    }
    __syncthreads();
    #pragma unroll 4
    for (int k0 = 0; k0 < 64; k0 += 4) {
      v2f a = *(const v2f*)(&AsBuf[0][(m0 + lhalf) * APITCH + k0 + kk]);
      const float* w0 = &WsBuf[0][(k0 + kk) * 128 + nbase + lhalf];
      const float* w1 = &WsBuf[0][(k0 + kk + 1) * 128 + nbase + lhalf];
      #pragma unroll
      for (int nt = 0; nt < 4; ++nt) {
        v2f bf; bf.x = w0[nt * 16]; bf.y = w1[nt * 16];
        acc[nt] = __builtin_amdgcn_wmma_f32_16x16x4_f32(
            false, a, false, bf, (short)0, acc[nt], false, false);
      }
    }
  }
#endif

  // write P = A@W into WsBuf[0] (64 x 128); D layout: VGPR v -> row m0+v(+8 hi)
  __syncthreads();
  const int ro = (lane >> 4) << 3;
  #pragma unroll
  for (int nt = 0; nt < 4; ++nt) {
    const int col = nbase + nt * 16 + lhalf;
    #pragma unroll
    for (int v = 0; v < 8; ++v)
      WsBuf[0][(m0 + ro + v) * 128 + col] = acc[nt][v];
  }
  __syncthreads();

  // stage 2: Z = M @ P, K = 64 (M rows: 16 KiB, cache-resident, aligned b64 loads)
  v8f zac[4] = {};
  #pragma unroll 4
  for (int k0 = 0; k0 < 64; k0 += 4) {
    v2f a = *(const v2f*)(Mg + (m0 + lhalf) * 64 + k0 + kk);
    const float* p0 = &WsBuf[0][(k0 + kk) * 128 + nbase + lhalf];
    const float* p1 = &WsBuf[0][(k0 + kk + 1) * 128 + nbase + lhalf];
    #pragma unroll
    for (int nt = 0; nt < 4; ++nt) {
      v2f bf; bf.x = p0[nt * 16]; bf.y = p1[nt * 16];
      zac[nt] = __builtin_amdgcn_wmma_f32_16x16x4_f32(
          false, a, false, bf, (short)0, zac[nt], false, false);
    }
  }

  // epilogue: + bias, store Z, accumulate BN partials (pre-activation stats)
  float* Zb = Zg + (size_t)b * 64 * HH;
  #pragma unroll
  for (int nt = 0; nt < 4; ++nt) {
    const int col = nbase + nt * 16 + lhalf;
    const float bv = bias[col];
    float s1 = 0.f, s2 = 0.f;
    #pragma unroll
    for (int v = 0; v < 8; ++v) {
      float val = zac[nt][v] + bv;
      Zb[(m0 + ro + v) * HH + col] = val;
      s1 += val; s2 += val * val;
    }
    s1 += __shfl_xor(s1, 16, 32);   // lane l and l^16 share the same column
    s2 += __shfl_xor(s2, 16, 32);
    if (lane < 16) { atomicAdd(&sSum[col], s1); atomicAdd(&sSqr[col], s2); }
  }
  __syncthreads();
  if (tid < HH) {
    psum[b * HH + tid] = sSum[tid];
    psqr[b * HH + tid] = sSqr[tid];
  }
}

// ---------------------------------------------------------------------------
// K7: BN stats over all N rows -> per-feature affine (a, c). Biased var (ddof=0).
__global__ void __launch_bounds__(128) k_bnstats(const float* __restrict__ psum,
                                                 const float* __restrict__ psqr,
                                                 const float* __restrict__ g,
                                                 const float* __restrict__ be,
                                                 float* __restrict__ af,
                                                 float* __restrict__ cf) {
  int f = threadIdx.x;
  float s = 0.f, q = 0.f;
  for (int i = 0; i < BB; ++i) { s += psum[i * HH + f]; q += psqr[i * HH + f]; }
  const float inv = 1.0f / (float)NN;
  float m   = s * inv;
  float var = q * inv - m * m;
  float a   = g[f] * rsqrtf(var + EPS_BN);
  af[f] = a;
  cf[f] = be[f] - m * a;
}

// K8: fused BN-affine + ReLU (elementwise, 4 elems/thread)
__global__ void __launch_bounds__(256) k_bnapply(const float* __restrict__ z,
                                                 const float* __restrict__ af,
                                                 const float* __restrict__ cf,
                                                 float* __restrict__ out) {
  int base = blockIdx.x * 1024 + threadIdx.x;
  #pragma unroll
  for (int j = 0; j < 4; ++j) {
    int idx = base + j * 256;
    int col = idx & (HH - 1);
    out[idx] = fmaxf(af[col] * z[idx] + cf[col], 0.f);
  }
}

// ---------------------------------------------------------------------------
extern "C" void kernel_launch(void* const* d_in, const int* in_sizes, int n_in,
                              void* d_out, int out_size, void* d_ws, size_t ws_size,
                              hipStream_t stream) {
  (void)in_sizes; (void)n_in; (void)out_size; (void)ws_size;
  const float* x   = (const float*)d_in[0];
  // d_in[1..8]: temporal-branch params (dead code in reference output) — unused
  const float* W1  = (const float*)d_in[9];
  const float* b1  = (const float*)d_in[10];
  const float* g1  = (const float*)d_in[11];
  const float* be1 = (const float*)d_in[12];
  const float* W2  = (const float*)d_in[13];
  const float* b2  = (const float*)d_in[14];
  const float* g2  = (const float*)d_in[15];
  const float* be2 = (const float*)d_in[16];
  const float* W3  = (const float*)d_in[17];
  const float* b3  = (const float*)d_in[18];
  const float* g3  = (const float*)d_in[19];
  const float* be3 = (const float*)d_in[20];
  float* out = (float*)d_out;

  char* w = (char*)d_ws;
  auto alloc = [&](size_t bytes) -> float* {
    float* p = (float*)w;
    w += (bytes + 255) & ~(size_t)255;
    return p;
  };
  float* nodes   = alloc((size_t)NN * TT * 4);   // 64 MiB
  float* zA      = alloc((size_t)NN * HH * 4);   // 16 MiB
  float* zB      = alloc((size_t)NN * HH * 4);   // 16 MiB
  float* rinv    = alloc((size_t)NN * 4);
  float* simpart = alloc((size_t)BB * 4096 * 4); // 8 MiB
  float* sim     = alloc(4096 * 4);
  float* Mmat    = alloc(4096 * 4);
  float* psum    = alloc((size_t)BB * HH * 4);
  float* psqr    = alloc((size_t)BB * HH * 4);
  float* af      = alloc(HH * 4);
  float* cf      = alloc(HH * 4);

  // graph construction
  k_rownorm  <<<NN / 8, 256, 0, stream>>>(x, rinv);
  k_nodes    <<<BB * (TT / 64), 256, 0, stream>>>(x, nodes);
  k_simpart  <<<BB, 256, 0, stream>>>(x, rinv, simpart);
  k_simreduce<<<16, 256, 0, stream>>>(simpart, sim);
  k_graph    <<<1, 64, 0, stream>>>(sim, Mmat);

  // layer 1: Z = M @ (nodes @ W1) + b1 ; BN ; ReLU
  k_gcn_gemm <<<BB, 256, 0, stream>>>(nodes, TT, W1, Mmat, b1, zA, psum, psqr);
  k_bnstats  <<<1, 128, 0, stream>>>(psum, psqr, g1, be1, af, cf);
  k_bnapply  <<<NN * HH / 1024, 256, 0, stream>>>(zA, af, cf, zA);

  // layer 2
  k_gcn_gemm <<<BB, 256, 0, stream>>>(zA, HH, W2, Mmat, b2, zB, psum, psqr);
  k_bnstats  <<<1, 128, 0, stream>>>(psum, psqr, g2, be2, af, cf);
  k_bnapply  <<<NN * HH / 1024, 256, 0, stream>>>(zB, af, cf, zB);

  // layer 3 -> d_out (flat (N,H) == (B, C*H))
  k_gcn_gemm <<<BB, 256, 0, stream>>>(zB, HH, W3, Mmat, b3, zA, psum, psqr);
  k_bnstats  <<<1, 128, 0, stream>>>(psum, psqr, g3, be3, af, cf);
  k_bnapply  <<<NN * HH / 1024, 256, 0, stream>>>(zA, af, cf, out);
}